// GCNBlock_41927470743917
// MI455X (gfx1250) — compile-verified
//
#include <hip/hip_runtime.h>
#include <math.h>

// GCNBlock (ViG Grapher + FFN), B=2048, C=32, N=144, K=9, fp32.
// One workgroup (256 thr = 8 wave32) per batch element; fully fused in LDS.
// All GEMMs on V_WMMA_F32_16X16X4_F32 (gfx1250 wave32 WMMA).

#define CC   32      // channels
#define NN   144     // H*W
#define KNN  9       // neighbors
#define THREADS 256
#define WAVES   8

typedef float v2f __attribute__((ext_vector_type(2)));
typedef float v8f __attribute__((ext_vector_type(8)));

// ---- WMMA 16x16x4 f32 fragment layout (CDNA5 ISA 7.12.2) ----
// A (16x4): lanes 0-15 -> rows, V0=K0,V1=K1 ; lanes 16-31 -> rows, V0=K2,V1=K3
// B (4x16): lanes 0-15 -> cols, V0=K0,V1=K1 ; lanes 16-31 -> cols, V0=K2,V1=K3
// C/D (16x16): VGPR j = row j (lanes 0-15) / row 8+j (lanes 16-31), lane&15 = col

// D-tile = W[mb:mb+16, :K] @ X[:K, nb:nb+16]; W row-major (ldw), X row-major (ldx)
__device__ __forceinline__ v8f tile_w_x(const float* __restrict__ W, int ldw,
                                        const float* X, int ldx,
                                        int mb, int nb, int K, int lane) {
  v8f acc = {0.f, 0.f, 0.f, 0.f, 0.f, 0.f, 0.f, 0.f};
  const int r  = lane & 15;
  const int kh = (lane >> 4) << 1;          // 0 or 2
  #pragma unroll
  for (int k0 = 0; k0 < 64; k0 += 4) {
    if (k0 >= K) break;
    v2f a, b;
    a.x = W[(mb + r) * ldw + (k0 + kh)];
    a.y = W[(mb + r) * ldw + (k0 + kh + 1)];
    b.x = X[(k0 + kh)     * ldx + nb + r];
    b.y = X[(k0 + kh + 1) * ldx + nb + r];
    acc = __builtin_amdgcn_wmma_f32_16x16x4_f32(false, a, false, b,
                                                (short)0, acc, false, false);
  }
  return acc;
}

// D-tile = X^T[mb:mb+16, :K] @ X[:K, nb:nb+16]  (Gram; A[m][k] = X[k][m])
__device__ __forceinline__ v8f tile_xt_x(const float* X, int ldx,
                                         int mb, int nb, int K, int lane) {
  v8f acc = {0.f, 0.f, 0.f, 0.f, 0.f, 0.f, 0.f, 0.f};
  const int r  = lane & 15;
  const int kh = (lane >> 4) << 1;
  #pragma unroll
  for (int k0 = 0; k0 < 32; k0 += 4) {
    if (k0 >= K) break;
    v2f a, b;
    a.x = X[(k0 + kh)     * ldx + mb + r];
    a.y = X[(k0 + kh + 1) * ldx + mb + r];
    b.x = X[(k0 + kh)     * ldx + nb + r];
    b.y = X[(k0 + kh + 1) * ldx + nb + r];
    acc = __builtin_amdgcn_wmma_f32_16x16x4_f32(false, a, false, b,
                                                (short)0, acc, false, false);
  }
  return acc;
}

__device__ __forceinline__ float gelu_exact(float x) {
  return 0.5f * x * (1.f + erff(x * 0.70710678118654752f));
}

// Dynamic LDS layout (floats):
//  xs  [32*144]  = 4608   : x = inputs + pos_embed (residual tmp; later FFN z)
//  xf  [32*144]  = 4608   : g_fc1 features (later y = grapher output)
//  uni [144*144] = 20736  : dist matrix, recycled as stacked[64*144] + hc[64*144]
//  inv [144], sq [144]
//  idx [144*9] ints
#define OFF_XS  0
#define OFF_XF  (OFF_XS + CC * NN)
#define OFF_UNI (OFF_XF + CC * NN)
#define OFF_INV (OFF_UNI + NN * NN)
#define OFF_SQ  (OFF_INV + NN)
#define OFF_IDX (OFF_SQ + NN)
#define SMEM_FLOATS (OFF_IDX + NN * KNN)
#define SMEM_BYTES  (SMEM_FLOATS * 4)

__global__ __launch_bounds__(THREADS)
void gcn_block_kernel(const float* __restrict__ g_in,
                      const float* __restrict__ pos,
                      const float* __restrict__ rel,
                      const float* __restrict__ w1,  const float* __restrict__ b1,
                      const float* __restrict__ wc,  const float* __restrict__ bc,
                      const float* __restrict__ w2,  const float* __restrict__ b2,
                      const float* __restrict__ wf1, const float* __restrict__ bf1,
                      const float* __restrict__ wf2, const float* __restrict__ bf2,
                      float* __restrict__ out) {
  extern __shared__ float smem[];
  float* xs  = smem + OFF_XS;
  float* xf  = smem + OFF_XF;
  float* uni = smem + OFF_UNI;
  float* inv = smem + OFF_INV;
  float* sq  = smem + OFF_SQ;
  int*   idx = (int*)(smem + OFF_IDX);

  const int tid  = threadIdx.x;
  const int wave = tid >> 5;
  const int lane = tid & 31;
  const int col16 = lane & 15;
  const int row8  = (lane >> 4) << 3;       // 0 or 8

  const float* xin = g_in + (size_t)blockIdx.x * (CC * NN);
  float*       og  = out  + (size_t)blockIdx.x * (CC * NN);

  // ---- x = inputs + pos_embed ----
  for (int i = tid; i < CC * NN; i += THREADS) xs[i] = xin[i] + pos[i];
  __syncthreads();

  // ---- x1 = g_fc1 @ x  (32x144, K=32): 18 tiles ----
  for (int t = wave; t < 18; t += WAVES) {
    const int mb = (t & 1) * 16, nb = (t >> 1) * 16;
    v8f acc = tile_w_x(w1, CC, xs, NN, mb, nb, CC, lane);
    #pragma unroll
    for (int j = 0; j < 8; ++j) {
      const int rr = mb + row8 + j;
      xf[rr * NN + nb + col16] = acc[j] + b1[rr];
    }
  }
  __syncthreads();

  // ---- column norms, sq = ||xn||^2 ----
  if (tid < NN) {
    float ss = 0.f;
    #pragma unroll
    for (int c = 0; c < CC; ++c) { float v = xf[c * NN + tid]; ss += v * v; }
    const float iv = 1.f / fmaxf(sqrtf(ss), 1e-12f);
    inv[tid] = iv;
    sq[tid]  = ss * iv * iv;
  }
  __syncthreads();

  // ---- dist = sq_n + sq_m - 2*Gram(xn)/.. + rel  (144x144, K=32): 81 tiles ----
  for (int t = wave; t < 81; t += WAVES) {
    const int mb = (t / 9) * 16, nb = (t % 9) * 16;
    v8f acc = tile_xt_x(xf, NN, mb, nb, CC, lane);
    const int cc = nb + col16;
    const float ivc = inv[cc], sqc = sq[cc];
    #pragma unroll
    for (int j = 0; j < 8; ++j) {
      const int rr = mb + row8 + j;
      uni[rr * NN + cc] = sq[rr] + sqc - 2.f * acc[j] * inv[rr] * ivc
                        + rel[rr * NN + cc];
    }
  }
  __syncthreads();

  // ---- top-9 smallest per row (strict-< insertion keeps first occurrence) ----
  if (tid < NN) {
    float bv[KNN]; int bi[KNN];
    #pragma unroll
    for (int k = 0; k < KNN; ++k) { bv[k] = 3.0e38f; bi[k] = 0; }
    const float* drow = uni + tid * NN;
    for (int j = 0; j < NN; ++j) {
      const float v = drow[j];
      if (v < bv[KNN - 1]) {
        int p = KNN - 1;
        while (p > 0 && v < bv[p - 1]) { bv[p] = bv[p - 1]; bi[p] = bi[p - 1]; --p; }
        bv[p] = v; bi[p] = j;
      }
    }
    #pragma unroll
    for (int k = 0; k < KNN; ++k) idx[tid * KNN + k] = bi[k];
  }
  __syncthreads();

  // ---- MRConv stack: stacked[2c]=xf[c], stacked[2c+1]=max_k xf[c][idx]-xf[c] ----
  float* stk = uni;             // 64*144 floats (recycle dist buffer)
  float* hc  = uni + 2 * CC * NN;
  for (int i = tid; i < CC * NN; i += THREADS) {
    const int c = i / NN, n = i - c * NN;
    const float xv = xf[i];
    const int* ip = idx + n * KNN;
    float mx = xf[c * NN + ip[0]];
    #pragma unroll
    for (int k = 1; k < KNN; ++k) mx = fmaxf(mx, xf[c * NN + ip[k]]);
    stk[(2 * c)     * NN + n] = xv;
    stk[(2 * c + 1) * NN + n] = mx - xv;
  }
  __syncthreads();

  // ---- hc = gelu(g_conv @ stacked + b)  (64x144, K=64): 36 tiles ----
  for (int t = wave; t < 36; t += WAVES) {
    const int mb = (t & 3) * 16, nb = (t >> 2) * 16;
    v8f acc = tile_w_x(wc, 2 * CC, stk, NN, mb, nb, 2 * CC, lane);
    #pragma unroll
    for (int j = 0; j < 8; ++j) {
      const int rr = mb + row8 + j;
      hc[rr * NN + nb + col16] = gelu_exact(acc[j] + bc[rr]);
    }
  }
  __syncthreads();

  // ---- y = g_fc2 @ hc + b + xs  -> xf  (32x144, K=64): 18 tiles ----
  for (int t = wave; t < 18; t += WAVES) {
    const int mb = (t & 1) * 16, nb = (t >> 1) * 16;
    v8f acc = tile_w_x(w2, 2 * CC, hc, NN, mb, nb, 2 * CC, lane);
    #pragma unroll
    for (int j = 0; j < 8; ++j) {
      const int rr = mb + row8 + j, cidx = rr * NN + nb + col16;
      xf[cidx] = acc[j] + b2[rr] + xs[cidx];
    }
  }
  __syncthreads();

  // ---- z = gelu(f_fc1 @ y + b)  -> xs  (32x144, K=32) ----
  for (int t = wave; t < 18; t += WAVES) {
    const int mb = (t & 1) * 16, nb = (t >> 1) * 16;
    v8f acc = tile_w_x(wf1, CC, xf, NN, mb, nb, CC, lane);
    #pragma unroll
    for (int j = 0; j < 8; ++j) {
      const int rr = mb + row8 + j;
      xs[rr * NN + nb + col16] = gelu_exact(acc[j] + bf1[rr]);
    }
  }
  __syncthreads();

  // ---- out = f_fc2 @ z + b + y  -> global ----
  for (int t = wave; t < 18; t += WAVES) {
    const int mb = (t & 1) * 16, nb = (t >> 1) * 16;
    v8f acc = tile_w_x(wf2, CC, xs, NN, mb, nb, CC, lane);
    #pragma unroll
    for (int j = 0; j < 8; ++j) {
      const int rr = mb + row8 + j, cidx = rr * NN + nb + col16;
      og[cidx] = acc[j] + bf2[rr] + xf[cidx];
    }
  }
}

extern "C" void kernel_launch(void* const* d_in, const int* in_sizes, int n_in,
                              void* d_out, int out_size, void* d_ws, size_t ws_size,
                              hipStream_t stream) {
  const float* inputs = (const float*)d_in[0];
  const float* pos    = (const float*)d_in[1];
  const float* rel    = (const float*)d_in[2];
  const float* w1  = (const float*)d_in[3];
  const float* b1  = (const float*)d_in[4];
  const float* wcv = (const float*)d_in[5];
  const float* bcv = (const float*)d_in[6];
  const float* w2  = (const float*)d_in[7];
  const float* b2  = (const float*)d_in[8];
  const float* wf1 = (const float*)d_in[9];
  const float* bf1 = (const float*)d_in[10];
  const float* wf2 = (const float*)d_in[11];
  const float* bf2 = (const float*)d_in[12];
  float* out = (float*)d_out;

  const int B = in_sizes[0] / (CC * NN);   // 2048

  (void)hipFuncSetAttribute((const void*)gcn_block_kernel,
                            hipFuncAttributeMaxDynamicSharedMemorySize,
                            SMEM_BYTES);
  gcn_block_kernel<<<B, THREADS, SMEM_BYTES, stream>>>(
      inputs, pos, rel, w1, b1, wcv, bcv, w2, b2, wf1, bf1, wf2, bf2, out);
}